// LocalSelfAttentionBase_24300924961595
// MI455X (gfx1250) — compile-verified
//
#include <hip/hip_runtime.h>
#include <hip/hip_bf16.h>

#define KVOL 27
#define BASE_ 130
#define BASE2 16900
#define BASE3 2197000
#define EMPTYK 0xFFFFFFFFu

typedef unsigned int u32x4 __attribute__((ext_vector_type(4)));
typedef int i32x8 __attribute__((ext_vector_type(8)));
typedef int i32x4 __attribute__((ext_vector_type(4)));

__device__ __forceinline__ unsigned hashSlot(unsigned k, unsigned mask) {
  k ^= k >> 16;
  k *= 0x9E3779B1u;
  k ^= k >> 13;
  return k & mask;
}

// Stage 256 points (768 dwords) of the coordinate stream into LDS.
// Preferred path: Tensor Data Mover (async DMA, TENSORcnt). Fallback: plain loads.
__device__ __forceinline__ void stageCoords(const float* __restrict__ coords,
                                            int N, int base, float* sc) {
#if __has_builtin(__builtin_amdgcn_tensor_load_to_lds) && __has_builtin(__builtin_amdgcn_s_wait_tensorcnt)
  if (threadIdx.x < 32u) {  // one wave issues the TDM DMA (EXEC ignored by TDM)
    unsigned long long ga = (unsigned long long)(const void*)(coords + (size_t)base * 3);
    unsigned ldsOff = (unsigned)(unsigned long long)(__attribute__((address_space(3))) float*)sc;
    unsigned tdim0 = (unsigned)N * 3u;  // full tensor length in dwords; OOB reads return 0
    u32x4 g0 = { 1u,                                        // count=1 (valid user descriptor)
                 ldsOff,                                    // lds_addr (bytes)
                 (unsigned)(ga & 0xFFFFFFFFull),            // global_addr[31:0]
                 (unsigned)((ga >> 32) & 0x1FFFFFFull) | (2u << 30) };  // addr[56:32] | type=2
    i32x8 g1 = { (int)(2u << 16),                           // data_size = 2 -> 4 bytes
                 (int)((tdim0 & 0xFFFFu) << 16),            // tensor_dim0[15:0] at bits 63:48
                 (int)((tdim0 >> 16) | (1u << 16)),         // tensor_dim0[31:16]; tensor_dim1 = 1
                 (int)(768u << 16),                         // tile_dim0 = 768 dwords (1D tile)
                 0, 0, 0, 0 };                              // tile_dim1/2 unused, strides 0
    i32x4 gz = { 0, 0, 0, 0 };
    i32x8 gz8 = { 0, 0, 0, 0, 0, 0, 0, 0 };
    __builtin_amdgcn_tensor_load_to_lds(g0, g1, gz, gz, gz8, 0);
    __builtin_amdgcn_s_wait_tensorcnt(0);
  }
  __syncthreads();
#else
  int i = base + (int)threadIdx.x;
  bool ok = i < N;
  size_t src = ok ? (size_t)i : 0;
  float a = coords[src * 3 + 0];
  float b = coords[src * 3 + 1];
  float c = coords[src * 3 + 2];
  sc[threadIdx.x * 3 + 0] = ok ? a : 0.f;
  sc[threadIdx.x * 3 + 1] = ok ? b : 0.f;
  sc[threadIdx.x * 3 + 2] = ok ? c : 0.f;
  __syncthreads();
#endif
}

__device__ __forceinline__ unsigned blockScanExcl(unsigned v, unsigned* lds, unsigned& total) {
  int tid = (int)threadIdx.x;
  lds[tid] = v;
  __syncthreads();
  for (int off = 1; off < 256; off <<= 1) {
    unsigned t = (tid >= off) ? lds[tid - off] : 0u;
    __syncthreads();
    lds[tid] += t;
    __syncthreads();
  }
  unsigned incl = lds[tid];
  total = lds[255];
  __syncthreads();
  return incl - v;
}

__global__ void fillU32Kernel(unsigned* __restrict__ p, unsigned v, long long n) {
  long long i = (long long)blockIdx.x * blockDim.x + threadIdx.x;
  long long stride = (long long)gridDim.x * blockDim.x;
  for (; i < n; i += stride) p[i] = v;
}

__global__ void batchStartsKernel(const int* __restrict__ batch, int N,
                                  unsigned* __restrict__ starts) {
  int i = (int)(blockIdx.x * blockDim.x + threadIdx.x);
  if (i < N) atomicMin(&starts[batch[i]], (unsigned)i);
}

// Phase 1: insert all T keys into the L2-resident open hash; record first position.
__global__ void buildHashKernel(const float* __restrict__ coords, const int* __restrict__ batch,
                                int N, unsigned capMask,
                                unsigned* __restrict__ hkey, unsigned* __restrict__ hfp) {
  __shared__ float sc[768];
  int base = (int)blockIdx.x * 256;
  int i = base + (int)threadIdx.x;
  if (i < N) __builtin_prefetch(&batch[i], 0, 3);  // global_prefetch_b8, overlaps TDM
  stageCoords(coords, N, base, sc);
  if (i >= N) return;
  int x = __float2int_rn(sc[threadIdx.x * 3 + 0]);
  int y = __float2int_rn(sc[threadIdx.x * 3 + 1]);
  int z = __float2int_rn(sc[threadIdx.x * 3 + 2]);
  int b = batch[i];
  unsigned p0 = (unsigned)(b * BASE3 + ((x + 1) * BASE_ + (y + 1)) * BASE_ + (z + 1));
  int t0 = i * KVOL;
  for (int k = 0; k < KVOL; ++k) {
    int ox = k / 9 - 1, oy = (k / 3) % 3 - 1, oz = k % 3 - 1;
    unsigned key = p0 + (unsigned)(ox * BASE2 + oy * BASE_ + oz);
    unsigned h = hashSlot(key, capMask);
    for (unsigned probe = 0; probe <= capMask; ++probe) {
      unsigned old = atomicCAS(&hkey[h], EMPTYK, key);
      if (old == EMPTYK || old == key) break;
      h = (h + 1) & capMask;
    }
    atomicMin(&hfp[h], (unsigned)(t0 + k));
  }
}

__global__ void setFlagsKernel(const unsigned* __restrict__ hkey, const unsigned* __restrict__ hfp,
                               unsigned* __restrict__ flags, unsigned cap) {
  unsigned s = blockIdx.x * 256u + threadIdx.x;
  if (s < cap && hkey[s] != EMPTYK) flags[hfp[s]] = 1u;
}

__global__ void scanPass1Kernel(const unsigned* __restrict__ flags, unsigned* __restrict__ bsums,
                                int T) {
  __shared__ unsigned lds[256];
  int tid = (int)threadIdx.x;
  int base = (int)blockIdx.x * 4096 + tid * 16;
  unsigned s = 0;
  for (int j = 0; j < 16; ++j) {
    int idx = base + j;
    if (idx < T) s += flags[idx];
  }
  lds[tid] = s;
  __syncthreads();
  for (int off = 128; off > 0; off >>= 1) {
    if (tid < off) lds[tid] += lds[tid + off];
    __syncthreads();
  }
  if (tid == 0) bsums[blockIdx.x] = lds[0];
}

__global__ void scanPass2Kernel(unsigned* __restrict__ bsums, int nb, int* __restrict__ numOut) {
  __shared__ unsigned lds[256];
  __shared__ unsigned carry;
  int tid = (int)threadIdx.x;
  if (tid == 0) carry = 0;
  __syncthreads();
  for (int baseI = 0; baseI < nb; baseI += 256) {
    int idx = baseI + tid;
    unsigned v = (idx < nb) ? bsums[idx] : 0u;
    unsigned total;
    unsigned ex = blockScanExcl(v, lds, total);
    unsigned c = carry;
    __syncthreads();
    if (idx < nb) bsums[idx] = ex + c;
    if (tid == 0) carry = c + total;
    __syncthreads();
  }
  if (tid == 0) {
    bsums[nb] = carry;
    *numOut = (int)carry;
  }
}

__global__ void scanPass3Kernel(unsigned* __restrict__ flags, const unsigned* __restrict__ bsums,
                                int T) {
  __shared__ unsigned lds[256];
  int tid = (int)threadIdx.x;
  int base = (int)blockIdx.x * 4096 + tid * 16;
  unsigned v[16];
  unsigned s = 0;
  for (int j = 0; j < 16; ++j) {
    int idx = base + j;
    unsigned f = (idx < T) ? flags[idx] : 0u;
    v[j] = s;
    s += f;
  }
  unsigned total;
  unsigned ex = blockScanExcl(s, lds, total);
  unsigned off = bsums[blockIdx.x] + ex;
  for (int j = 0; j < 16; ++j) {
    int idx = base + j;
    if (idx < T) flags[idx] = v[j] + off;  // in place: flags -> exclusive ranks
  }
}

// Scatter unique output coordinates in insertion order.
__global__ void emitKeysKernel(const unsigned* __restrict__ hkey, const unsigned* __restrict__ hfp,
                               const unsigned* __restrict__ ranks, const float* __restrict__ coords,
                               int* __restrict__ outKey, unsigned cap) {
  unsigned s = blockIdx.x * 256u + threadIdx.x;
  if (s >= cap) return;
  if (hkey[s] == EMPTYK) return;
  unsigned fp = hfp[s];
  unsigned r = ranks[fp];
  int i = (int)(fp / KVOL), k = (int)(fp % KVOL);
  int ox = k / 9 - 1, oy = (k / 3) % 3 - 1, oz = k % 3 - 1;
  outKey[(size_t)r * 3 + 0] = __float2int_rn(coords[(size_t)i * 3 + 0]) + ox;
  outKey[(size_t)r * 3 + 1] = __float2int_rn(coords[(size_t)i * 3 + 1]) + oy;
  outKey[(size_t)r * 3 + 2] = __float2int_rn(coords[(size_t)i * 3 + 2]) + oz;
}

// Final per-entry outputs: in_idx, out_idx, rel_idx (re-probe find, table is L2-resident).
__global__ void writeOutputsKernel(const float* __restrict__ coords, const int* __restrict__ batch,
                                   int N, unsigned capMask,
                                   const unsigned* __restrict__ hkey,
                                   const unsigned* __restrict__ hfp,
                                   const unsigned* __restrict__ ranks,
                                   const unsigned* __restrict__ starts,
                                   int* __restrict__ out, int T) {
  __shared__ float sc[768];
  int base = (int)blockIdx.x * 256;
  int i = base + (int)threadIdx.x;
  if (i < N) __builtin_prefetch(&batch[i], 0, 3);
  stageCoords(coords, N, base, sc);
  if (i >= N) return;
  int x = __float2int_rn(sc[threadIdx.x * 3 + 0]);
  int y = __float2int_rn(sc[threadIdx.x * 3 + 1]);
  int z = __float2int_rn(sc[threadIdx.x * 3 + 2]);
  int b = batch[i];
  int local = i - (int)starts[b];
  unsigned p0 = (unsigned)(b * BASE3 + ((x + 1) * BASE_ + (y + 1)) * BASE_ + (z + 1));
  int t0 = i * KVOL;
  for (int k = 0; k < KVOL; ++k) {
    int ox = k / 9 - 1, oy = (k / 3) % 3 - 1, oz = k % 3 - 1;
    unsigned key = p0 + (unsigned)(ox * BASE2 + oy * BASE_ + oz);
    unsigned h = hashSlot(key, capMask);
    unsigned probe = 0;
    while (hkey[h] != key && probe++ <= capMask) h = (h + 1) & capMask;
    unsigned r = ranks[hfp[h]];
    int t = t0 + k;
    out[t] = local;                      // in_idx
    out[(size_t)T + t] = (int)r;         // out_idx
    out[2 * (size_t)T + t] = k;          // rel_idx
  }
}

extern "C" void kernel_launch(void* const* d_in, const int* in_sizes, int n_in,
                              void* d_out, int out_size, void* d_ws, size_t ws_size,
                              hipStream_t stream) {
  const float* coords = (const float*)d_in[0];
  const int* batch = (const int*)d_in[1];
  int N = in_sizes[1];
  int T = N * KVOL;

  // Hash capacity: power of two >= 1.5*T (load factor <= 0.67). For N=200k -> 2^23 slots,
  // 64 MB of tables -> fully resident in the 192 MB L2.
  unsigned cap = 1024;
  while (cap < (unsigned)T + ((unsigned)T >> 1)) cap <<= 1;
  unsigned capMask = cap - 1;
  int NSB = (T + 4095) / 4096;

  char* ws = (char*)d_ws;
  size_t off = 0;
  auto carve = [&](size_t bytes) {
    char* p = ws + off;
    off = (off + bytes + 255) & ~(size_t)255;
    return p;
  };
  unsigned* hkey = (unsigned*)carve((size_t)cap * 4);
  unsigned* hfp = (unsigned*)carve((size_t)cap * 4);
  unsigned* flags = (unsigned*)carve((size_t)T * 4);        // later: exclusive ranks
  unsigned* bsums = (unsigned*)carve((size_t)(NSB + 1) * 4);
  unsigned* starts = (unsigned*)carve(8 * 4);
  (void)ws_size;
  (void)n_in;
  (void)out_size;

  int* out = (int*)d_out;
  int PB = (N + 255) / 256;
  int CB = (int)((cap + 255u) / 256u);

  fillU32Kernel<<<2048, 256, 0, stream>>>(hkey, EMPTYK, (long long)cap);
  fillU32Kernel<<<2048, 256, 0, stream>>>(hfp, EMPTYK, (long long)cap);
  fillU32Kernel<<<2048, 256, 0, stream>>>(flags, 0u, (long long)T);
  fillU32Kernel<<<1, 32, 0, stream>>>(starts, (unsigned)N, 8);
  fillU32Kernel<<<2048, 256, 0, stream>>>((unsigned*)(out + 3 * (size_t)T), EMPTYK,
                                          3LL * (long long)T);  // out_key = -1 padding

  batchStartsKernel<<<PB, 256, 0, stream>>>(batch, N, starts);
  buildHashKernel<<<PB, 256, 0, stream>>>(coords, batch, N, capMask, hkey, hfp);
  setFlagsKernel<<<CB, 256, 0, stream>>>(hkey, hfp, flags, cap);
  scanPass1Kernel<<<NSB, 256, 0, stream>>>(flags, bsums, T);
  scanPass2Kernel<<<1, 256, 0, stream>>>(bsums, NSB, out + 6 * (size_t)T);  // num_out
  scanPass3Kernel<<<NSB, 256, 0, stream>>>(flags, bsums, T);
  emitKeysKernel<<<CB, 256, 0, stream>>>(hkey, hfp, flags, coords, out + 3 * (size_t)T, cap);
  writeOutputsKernel<<<PB, 256, 0, stream>>>(coords, batch, N, capMask, hkey, hfp, flags,
                                             starts, out, T);
}